// TGMandTRM_11673721110797
// MI455X (gfx1250) — compile-verified
//
#include <hip/hip_runtime.h>
#include <math.h>

// Problem constants (match reference)
#define B_  32
#define C_  512
#define H_  64
#define R_  4
#define HW  (H_ * H_)     // 4096 elements per (b,c) slab
#define NC  (R_ * C_)     // 2048 = gC columns
#define NH  (R_ * H_)     // 256  = gH/gW columns

typedef __attribute__((ext_vector_type(2))) float v2f;
typedef __attribute__((ext_vector_type(8))) float v8f;

// ---------------------------------------------------------------------------
// Kernel 0: zero the Hv/Wv atomic accumulators (workspace is poisoned 0xAA)
// ---------------------------------------------------------------------------
__global__ void tgm_zero_kernel(float* __restrict__ p, int n) {
    int i = blockIdx.x * blockDim.x + threadIdx.x;
    if (i < n) p[i] = 0.0f;
}

// ---------------------------------------------------------------------------
// Kernel 1: pooled sums. One 256-thread block per (b,c) slab of 64x64 floats.
//   Cv[b,c]  = sum over (h,w)          (owned by block -> direct store)
//   Wv[b,h] += sum over w (row sums)   (global fp32 atomic across c-blocks)
//   Hv[b,w] += sum over h (col sums)   (global fp32 atomic across c-blocks)
// Means are recovered later via the gate-GEMM scale factor.
// ---------------------------------------------------------------------------
__global__ void tgm_reduce_kernel(const float* __restrict__ x,
                                  float* __restrict__ Cv,
                                  float* __restrict__ Hv,
                                  float* __restrict__ Wv) {
    const int bc = blockIdx.x;            // b*512 + c
    const int b  = bc >> 9;
    const int t  = threadIdx.x;           // 0..255

    __shared__ float rowS[H_];            // indexed by h  -> Wv
    __shared__ float colS[H_];            // indexed by w  -> Hv

    if (t < H_) { rowS[t] = 0.0f; colS[t] = 0.0f; }
    __syncthreads();

    const float4* xs = (const float4*)(x + (size_t)bc * HW);

    // Thread t always touches columns w..w+3 (invariant over i) -> registers.
    const int w = 4 * (t & 15);
    float4 colacc = make_float4(0.f, 0.f, 0.f, 0.f);

#pragma unroll
    for (int i = 0; i < 4; ++i) {
        const float4 v = xs[i * 256 + t];       // coalesced 16B per lane
        const int h = i * 16 + (t >> 4);
        atomicAdd(&rowS[h], v.x + v.y + v.z + v.w);   // ds_add_f32
        colacc.x += v.x; colacc.y += v.y; colacc.z += v.z; colacc.w += v.w;
    }
    atomicAdd(&colS[w + 0], colacc.x);
    atomicAdd(&colS[w + 1], colacc.y);
    atomicAdd(&colS[w + 2], colacc.z);
    atomicAdd(&colS[w + 3], colacc.w);
    __syncthreads();

    if (t < H_) {
        atomicAdd(&Wv[b * H_ + t], rowS[t]);              // global_atomic_add_f32
    } else if (t < 2 * H_) {
        atomicAdd(&Hv[b * H_ + (t - H_)], colS[t - H_]);
    } else if (t == 2 * H_) {
        float s = 0.0f;
#pragma unroll
        for (int h = 0; h < H_; ++h) s += rowS[h];        // total = sum of row sums
        Cv[bc] = s;
    }
}

// ---------------------------------------------------------------------------
// Kernel 2: gate GEMM + sigmoid using V_WMMA_F32_16X16X4_F32.
//   G (32 x N) = sigmoid( (A * scale) (32 x K)  @  Wf^T (K x N) )
//   Wf is (N, K) row-major (the flattened (rank, out, in) weight).
// One wave (32 threads) per 16x16 output tile; EXEC is all-ones as required.
// A-frag layout : lane(0..15)=M, half-wave selects K pair {k0, k0+1}/{k0+2,k0+3}
// B-frag layout : lane(0..15)=N, same K-pair split
// D layout      : vgpr v -> M = v + 8*(lane>=16), N = lane&15
// ---------------------------------------------------------------------------
__global__ void tgm_gate_wmma_kernel(const float* __restrict__ A,   // (32, K) sums
                                     const float* __restrict__ Wf,  // (N, K)
                                     float* __restrict__ G,         // (32, N)
                                     int K, int N, float scale) {
    const int l     = threadIdx.x;        // 0..31, full wave
    const int tileN = blockIdx.x;
    const int tileM = blockIdx.y;
    const int half  = l >> 4;             // 0: K pair {0,1}; 1: K pair {2,3}
    const int lane  = l & 15;

    const int m = tileM * 16 + lane;      // batch row for A fragment
    const int n = tileN * 16 + lane;      // output col for B fragment / store

    const float* arow = A  + (size_t)m * K;
    const float* brow = Wf + (size_t)n * K;

    v8f acc = {0.f, 0.f, 0.f, 0.f, 0.f, 0.f, 0.f, 0.f};

    for (int k0 = 0; k0 < K; k0 += 4) {
        const int kk = k0 + 2 * half;     // even -> 8B aligned
        v2f a;
        a.x = arow[kk]     * scale;
        a.y = arow[kk + 1] * scale;
        v2f bfrag;
        bfrag.x = brow[kk];
        bfrag.y = brow[kk + 1];
        // D = A(16x4) x B(4x16) + C ; fp32 throughout (matches reference dtype)
        acc = __builtin_amdgcn_wmma_f32_16x16x4_f32(
            /*neg_a=*/false, a, /*neg_b=*/false, bfrag,
            /*c_mod=*/(short)0, acc, /*reuse_a=*/false, /*reuse_b=*/false);
    }

    const int outm = tileM * 16 + 8 * half;
#pragma unroll
    for (int v = 0; v < 8; ++v) {
        const float d = acc[v];
        G[(size_t)(outm + v) * N + n] = 1.0f / (1.0f + expf(-d));   // sigmoid
    }
}

// ---------------------------------------------------------------------------
// Kernel 3: out = x + relu(x * T),  T[b,c,h,w] = sum_r a_r * gH[b,r,h] * gW[b,r,w]
// with a_r = softmax(lam)[r] * gC[b,r,c].  One block per (b,c) slab; gH/gW rows
// staged in LDS; float4 streaming of x -> out (pure HBM-bound pass).
// ---------------------------------------------------------------------------
__global__ void tgm_apply_kernel(const float* __restrict__ x,
                                 const float* __restrict__ gC,
                                 const float* __restrict__ gH,
                                 const float* __restrict__ gW,
                                 const float* __restrict__ lam,
                                 float* __restrict__ out) {
    const int bc = blockIdx.x;
    const int b  = bc >> 9;
    const int c  = bc & (C_ - 1);
    const int t  = threadIdx.x;

    __shared__ __align__(16) float GHs[NH];
    __shared__ __align__(16) float GWs[NH];
    __shared__ float aS[R_];

    GHs[t] = gH[(size_t)b * NH + t];
    GWs[t] = gW[(size_t)b * NH + t];
    if (t < R_) {
        const float l0 = lam[0], l1 = lam[1], l2 = lam[2], l3 = lam[3];
        const float mx = fmaxf(fmaxf(l0, l1), fmaxf(l2, l3));
        const float e0 = expf(l0 - mx), e1 = expf(l1 - mx);
        const float e2 = expf(l2 - mx), e3 = expf(l3 - mx);
        const float inv = 1.0f / (e0 + e1 + e2 + e3);
        const float er = (t == 0) ? e0 : (t == 1) ? e1 : (t == 2) ? e2 : e3;
        aS[t] = er * inv * gC[(size_t)b * NC + t * C_ + c];
    }
    __syncthreads();

    const float a0 = aS[0], a1 = aS[1], a2 = aS[2], a3 = aS[3];

    // Per-thread column window w..w+3 is loop-invariant: hoist gW reads.
    const int w = 4 * (t & 15);
    const float4 g0 = *(const float4*)&GWs[0 * H_ + w];
    const float4 g1 = *(const float4*)&GWs[1 * H_ + w];
    const float4 g2 = *(const float4*)&GWs[2 * H_ + w];
    const float4 g3 = *(const float4*)&GWs[3 * H_ + w];

    const float4* xs = (const float4*)(x   + (size_t)bc * HW);
    float4*       os = (float4*)      (out + (size_t)bc * HW);

#pragma unroll
    for (int i = 0; i < 4; ++i) {
        const int h = i * 16 + (t >> 4);
        const float p0 = a0 * GHs[0 * H_ + h];
        const float p1 = a1 * GHs[1 * H_ + h];
        const float p2 = a2 * GHs[2 * H_ + h];
        const float p3 = a3 * GHs[3 * H_ + h];

        const float4 v = xs[i * 256 + t];
        float4 T, y;
        T.x = p0 * g0.x + p1 * g1.x + p2 * g2.x + p3 * g3.x;
        T.y = p0 * g0.y + p1 * g1.y + p2 * g2.y + p3 * g3.y;
        T.z = p0 * g0.z + p1 * g1.z + p2 * g2.z + p3 * g3.z;
        T.w = p0 * g0.w + p1 * g1.w + p2 * g2.w + p3 * g3.w;
        y.x = v.x + fmaxf(v.x * T.x, 0.0f);
        y.y = v.y + fmaxf(v.y * T.y, 0.0f);
        y.z = v.z + fmaxf(v.z * T.z, 0.0f);
        y.w = v.w + fmaxf(v.w * T.w, 0.0f);
        os[i * 256 + t] = y;
    }
}

// ---------------------------------------------------------------------------
// Host launch
// ---------------------------------------------------------------------------
extern "C" void kernel_launch(void* const* d_in, const int* in_sizes, int n_in,
                              void* d_out, int out_size, void* d_ws, size_t ws_size,
                              hipStream_t stream) {
    const float* x   = (const float*)d_in[0];  // (32,512,64,64)
    const float* W1  = (const float*)d_in[1];  // (4,512,512)  -> flat (2048,512)
    const float* W2  = (const float*)d_in[2];  // (4,64,64)    -> flat (256,64)
    const float* W3  = (const float*)d_in[3];  // (4,64,64)
    const float* lam = (const float*)d_in[4];  // (4,)
    float* out = (float*)d_out;

    // Workspace layout (floats): total 102400 floats = 400 KB
    float* ws = (float*)d_ws;
    float* Cv = ws;                  // 32*512  = 16384
    float* Hv = ws + 16384;          // 32*64   = 2048
    float* Wv = ws + 18432;          // 32*64   = 2048
    float* gC = ws + 20480;          // 32*2048 = 65536
    float* gH = ws + 86016;          // 32*256  = 8192
    float* gW = ws + 94208;          // 32*256  = 8192

    // 0) zero Hv+Wv accumulators (contiguous 4096 floats)
    tgm_zero_kernel<<<16, 256, 0, stream>>>(Hv, 2 * B_ * H_);

    // 1) pooled sums
    tgm_reduce_kernel<<<B_ * C_, 256, 0, stream>>>(x, Cv, Hv, Wv);

    // 2) gate GEMMs (WMMA fp32) + sigmoid; scale folds the mean divisor
    dim3 gC_grid(NC / 16, B_ / 16);   // 128 x 2 tiles
    dim3 gH_grid(NH / 16, B_ / 16);   // 16 x 2 tiles
    tgm_gate_wmma_kernel<<<gC_grid, 32, 0, stream>>>(Cv, W1, gC, C_, NC, 1.0f / (float)HW);
    tgm_gate_wmma_kernel<<<gH_grid, 32, 0, stream>>>(Hv, W2, gH, H_, NH, 1.0f / (float)(C_ * H_));
    tgm_gate_wmma_kernel<<<gH_grid, 32, 0, stream>>>(Wv, W3, gW, H_, NH, 1.0f / (float)(C_ * H_));

    // 3) streaming apply (bandwidth-bound: 512 MB of traffic @ 23.3 TB/s)
    tgm_apply_kernel<<<B_ * C_, 256, 0, stream>>>(x, gC, gH, gW, lam, out);
}